// RoIInterpolationPool_19971597926911
// MI455X (gfx1250) — compile-verified
//
#include <hip/hip_runtime.h>

// RoI crop + bilinear resize (align_corners) to 14x14, fused, fp32.
// Input  d_in[0]: (B=2, C=256, H=256, W=256) float
// Input  d_in[1]: (N, 5) float rois  [bidx, x1, y1, x2, y2]
// Output d_out  : (N, 256, 14, 14) float
//
// One wave per (roi, 16-channel block). x-interpolation is done as a chain of
// V_WMMA_F32_16X16X4_F32:  D[16c x 16ox] += A[16c x 4k] * B[4k x 16ox]
//   A = y-blended image columns (per-lane channel row, K-pair in 2 VGPRs)
//   B = bilinear x-weights (per-lane ox column, K-pair in 2 VGPRs)
// Full fp32 precision; input (134MB) fits in the 192MB L2 so gathers are
// L2-resident; stores are lane-contiguous along ox.

typedef __attribute__((ext_vector_type(2))) float v2f;
typedef __attribute__((ext_vector_type(8))) float v8f;

#define C_TOT 256
#define H_IN  256
#define W_IN  256
#define OHW   14

static __device__ __forceinline__ int imind(int a, int b) { return a < b ? a : b; }
static __device__ __forceinline__ int imaxd(int a, int b) { return a > b ? a : b; }

__global__ __launch_bounds__(128) void roi_interp_pool_wmma(
    const float* __restrict__ inp, const float* __restrict__ rois,
    float* __restrict__ out, int nroi)
{
  const int lane = threadIdx.x & 31;
  const int wave = threadIdx.x >> 5;          // 0..3
  const int cblk = blockIdx.x * 4 + wave;     // 0..15  (16 channels each)
  const int roi  = blockIdx.y;
  const int cbase = cblk * 16;
  const int half = lane >> 4;                 // 0: K 0,1 | 1: K 2,3
  const int lid  = lane & 15;                 // A row (channel) and B col (ox)

  // ---- ROI parameters (wave-uniform) ----
  const float* rp = rois + (size_t)roi * 5;
  const int   b  = (int)rp[0];
  const float x1 = floorf(rp[1] * 0.25f);
  const float y1 = floorf(rp[2] * 0.25f);
  const float x2 = ceilf (rp[3] * 0.25f);
  const float y2 = ceilf (rp[4] * 0.25f);

  // x span of all taps: xlo = x0(ox=0), xhi = x1i(ox=13)
  int xlo = imind(imaxd((int)x1, 0), W_IN - 1);
  int xhi = imind(imaxd((int)floorf(x2 - 1.0f), 0), W_IN - 1) + 1;
  xhi = imind(xhi, W_IN - 1);
  const int kmax = xhi - xlo;                 // K-1, K <= ~34

  // ---- per-lane x-weights for ox = lid (branchless; ox>=14 -> zero weights) ----
  const int   oxc = imind(lid, OHW - 1);
  const float xs  = x1 + ((float)oxc / 13.0f) * (x2 - x1 - 1.0f);
  const float x0f = floorf(xs);
  const float fx  = xs - x0f;
  int x0i = imind(imaxd((int)x0f, 0), W_IN - 1);
  int x1i = imind(imaxd(x0i + 1, 0), W_IN - 1);
  const bool oxvalid = (lid < OHW);
  const int ix0 = oxvalid ? (x0i - xlo) : -1000000;
  const int ix1 = oxvalid ? (x1i - xlo) : -1000000;

  // ---- per-lane channel base pointer for A rows ----
  const float* chbase =
      inp + ((size_t)b * C_TOT + (size_t)(cbase + lid)) * ((size_t)H_IN * W_IN);

  for (int oy = 0; oy < OHW; ++oy) {
    // y taps (wave-uniform)
    const float ysv = y1 + ((float)oy / 13.0f) * (y2 - y1 - 1.0f);
    const float y0f = floorf(ysv);
    const float fy  = ysv - y0f;
    const int y0  = imind(imaxd((int)y0f, 0), H_IN - 1);
    const int y1r = imind(imaxd(y0 + 1, 0), H_IN - 1);
    const float wy0 = 1.0f - fy;
    const float* rowT = chbase + (size_t)y0  * W_IN;
    const float* rowB = chbase + (size_t)y1r * W_IN;

    v8f acc = {};   // C/D accumulator, 16x16 fp32

    for (int k0 = 0; k0 <= kmax; k0 += 4) {
      const int ka = k0 + (half ? 2 : 0);
      const int kb = ka + 1;
      const int xa = imind(xlo + ka, W_IN - 1);
      const int xb = imind(xlo + kb, W_IN - 1);

      // A: y-blended image, row = channel (lid), K pair in .x/.y
      v2f A;
      A.x = wy0 * rowT[xa] + fy * rowB[xa];
      A.y = wy0 * rowT[xb] + fy * rowB[xb];

      // B: x-interp weights, col = ox (lid), K pair in .x/.y
      v2f Bv;
      Bv.x = (ka == ix0 ? 1.0f - fx : 0.0f) + (ka == ix1 ? fx : 0.0f);
      Bv.y = (kb == ix0 ? 1.0f - fx : 0.0f) + (kb == ix1 ? fx : 0.0f);

      // D = A x B + C   (V_WMMA_F32_16X16X4_F32, wave32)
      acc = __builtin_amdgcn_wmma_f32_16x16x4_f32(
          /*neg_a=*/false, A, /*neg_b=*/false, Bv,
          /*c_mod=*/(short)0, acc, /*reuse_a=*/false, /*reuse_b=*/false);
    }

    // Store D: lane -> column ox = lid; VGPR r -> channel cbase + r + 8*half.
    // Lanes 0..13 of each half write 14 contiguous floats per r (coalesced).
    if (oxvalid) {
      const size_t obase =
          (((size_t)roi * C_TOT + (size_t)(cbase + 8 * half)) * OHW + oy) * OHW + lid;
      #pragma unroll
      for (int r = 0; r < 8; ++r)
        out[obase + (size_t)r * (OHW * OHW)] = acc[r];
    }
  }
}

extern "C" void kernel_launch(void* const* d_in, const int* in_sizes, int n_in,
                              void* d_out, int out_size, void* d_ws, size_t ws_size,
                              hipStream_t stream) {
  const float* inp  = (const float*)d_in[0];   // (2,256,256,256) f32
  const float* rois = (const float*)d_in[1];   // (N,5) f32
  float* out = (float*)d_out;                  // (N,256,14,14) f32
  const int nroi = in_sizes[1] / 5;

  dim3 grid(4, nroi);          // 4 waves/block * 4 blocks.x = 16 channel-blocks
  dim3 block(128);             // 4 wave32 per block
  roi_interp_pool_wmma<<<grid, block, 0, stream>>>(inp, rois, out, nroi);
}